// GraphTransformerBaseBlock_27152783245688
// MI455X (gfx1250) — compile-verified
//
#include <hip/hip_runtime.h>
#include <math.h>

typedef float v2f __attribute__((ext_vector_type(2)));
typedef float v8f __attribute__((ext_vector_type(8)));

// ---------------- helpers ----------------

__device__ __forceinline__ float gelu_tanh(float x) {
    const float k0 = 0.7978845608028654f;   // sqrt(2/pi)
    const float k1 = 0.044715f;
    float x3 = x * x * x;
    return 0.5f * x * (1.0f + tanhf(k0 * (x + k1 * x3)));
}

// fp32 atomic max via int/uint ordering trick (works for all finite + -inf init)
__device__ __forceinline__ void atomicMaxF(float* addr, float val) {
    if (val >= 0.0f) atomicMax((int*)addr, __float_as_int(val));
    else             atomicMin((unsigned int*)addr, __float_as_uint(val));
}

// ---------------- elementwise utility kernels ----------------

__global__ void fill_kernel(float* __restrict__ p, int n, float v) {
    int i = blockIdx.x * blockDim.x + threadIdx.x;
    if (i < n) p[i] = v;
}

__global__ void copy_kernel(const float* __restrict__ src, float* __restrict__ dst, int n) {
    int i = blockIdx.x * blockDim.x + threadIdx.x;
    if (i < n) dst[i] = src[i];
}

__global__ void add2_kernel(const float* __restrict__ a, const float* __restrict__ b,
                            float* __restrict__ c, int n) {
    int i = blockIdx.x * blockDim.x + threadIdx.x;
    if (i < n) c[i] = a[i] + b[i];
}

__global__ void fix_smax_kernel(float* __restrict__ s, int n) {
    int i = blockIdx.x * blockDim.x + threadIdx.x;
    if (i < n) {
        float v = s[i];
        if (!isfinite(v)) s[i] = 0.0f;   // empty-segment guard (matches reference)
    }
}

// ---------------- LayerNorm over 128 features: one wave per row ----------------

__global__ __launch_bounds__(256) void ln128_kernel(const float* __restrict__ x,
                                                    const float* __restrict__ w,
                                                    const float* __restrict__ b,
                                                    float* __restrict__ y, int nrows) {
    int lane = threadIdx.x & 31;
    int row  = blockIdx.x * (blockDim.x >> 5) + (threadIdx.x >> 5);
    if (row >= nrows) return;
    int c = lane * 4;
    float4 xv = *(const float4*)(x + (size_t)row * 128 + c);
    float s = xv.x + xv.y + xv.z + xv.w;
#pragma unroll
    for (int o = 1; o < 32; o <<= 1) s += __shfl_xor(s, o, 32);
    float mean = s * (1.0f / 128.0f);
    float dx = xv.x - mean, dy = xv.y - mean, dz = xv.z - mean, dw = xv.w - mean;
    float q = dx * dx + dy * dy + dz * dz + dw * dw;
#pragma unroll
    for (int o = 1; o < 32; o <<= 1) q += __shfl_xor(q, o, 32);
    float inv = rsqrtf(q * (1.0f / 128.0f) + 1e-5f);
    float4 wv = *(const float4*)(w + c);
    float4 bv = *(const float4*)(b + c);
    float4 out;
    out.x = dx * inv * wv.x + bv.x;
    out.y = dy * inv * wv.y + bv.y;
    out.z = dz * inv * wv.z + bv.z;
    out.w = dw * inv * wv.w + bv.w;
    *(float4*)(y + (size_t)row * 128 + c) = out;
}

// ---------------- fp32 WMMA GEMM: C[M x NC] = A[M x K] @ W[K x NC] + bias ----------------
// One wave computes one 16x16 C tile with V_WMMA_F32_16X16X4_F32 over the K loop.
// MODE 0: plain; MODE 1: + residual R; MODE 2: tanh-GELU.
// A strip (16 x K) staged in LDS with +1 padding -> conflict-free A reads.

template <int MODE>
__global__ __launch_bounds__(256) void gemm_wmma(const float* __restrict__ A,
                                                 const float* __restrict__ W,
                                                 const float* __restrict__ bias,
                                                 const float* __restrict__ R,
                                                 float* __restrict__ C, int K, int NC) {
    extern __shared__ float smem[];
    const int KP = K + 1;
    const int rowBase = blockIdx.x * 16;
    const int K4 = K >> 2;
    for (int t = threadIdx.x; t < 16 * K4; t += blockDim.x) {
        int r = t / K4, c4 = (t - r * K4) * 4;
        float4 av = *(const float4*)(A + (size_t)(rowBase + r) * K + c4);
        float* dp = smem + r * KP + c4;
        dp[0] = av.x; dp[1] = av.y; dp[2] = av.z; dp[3] = av.w;
    }
    __syncthreads();

    const int lane = threadIdx.x & 31;
    const int wv   = threadIdx.x >> 5;
    const int m    = lane & 15;     // A row within tile
    const int hi   = lane >> 4;     // half-wave selects K pair (A) / row pair (B/C)
    const int n    = lane & 15;     // B/C column within tile
    const int nct  = NC >> 4;

    for (int ct = wv; ct < nct; ct += 8) {
        const int colBase = ct * 16;
        v8f acc = {};
        const float* wp = W + colBase + n;
        for (int kk = 0; kk < K; kk += 4) {
            v2f a, bb;
            a.x  = smem[m * KP + kk + 2 * hi];
            a.y  = smem[m * KP + kk + 2 * hi + 1];
            bb.x = wp[(size_t)(kk + 2 * hi) * NC];
            bb.y = wp[(size_t)(kk + 2 * hi + 1) * NC];
            acc = __builtin_amdgcn_wmma_f32_16x16x4_f32(false, a, false, bb,
                                                        (short)0, acc, false, false);
        }
        const float bv = bias[colBase + n];
#pragma unroll
        for (int g = 0; g < 8; ++g) {
            int row = rowBase + g + 8 * hi;        // C layout: VGPR g -> row g + 8*hi
            size_t idx = (size_t)row * NC + colBase + n;
            float val = acc[g] + bv;
            if (MODE == 1) val += R[idx];
            if (MODE == 2) val = gelu_tanh(val);
            C[idx] = val;
        }
    }
}

// ---------------- edge kernels: one wave32 per edge, 4 channels per lane ----------------
// e = edge_attr @ We + be recomputed on the fly from LDS-resident We (8 KB).

__global__ __launch_bounds__(256) void edge_score_kernel(const float* __restrict__ q,
                                                         const float* __restrict__ k,
                                                         const float* __restrict__ edge_attr,
                                                         const float* __restrict__ We,
                                                         const float* __restrict__ be,
                                                         const int* __restrict__ ei,
                                                         float* __restrict__ scores,
                                                         float* __restrict__ smax, int E) {
    __shared__ float sWe[16 * 128];
    __shared__ float sbe[128];
    for (int t = threadIdx.x; t < 2048; t += blockDim.x) sWe[t] = We[t];
    for (int t = threadIdx.x; t < 128; t += blockDim.x) sbe[t] = be[t];
    __syncthreads();
    int lane = threadIdx.x & 31;
    int e = blockIdx.x * 8 + (threadIdx.x >> 5);
    if (e >= E) return;
    int s = ei[e], d = ei[E + e];
    float ea = (lane < 16) ? edge_attr[(size_t)e * 16 + lane] : 0.0f;
    int c0 = lane * 4;
    float4 qv = *(const float4*)(q + (size_t)d * 128 + c0);
    float4 kv = *(const float4*)(k + (size_t)s * 128 + c0);
    float e0 = sbe[c0], e1 = sbe[c0 + 1], e2 = sbe[c0 + 2], e3 = sbe[c0 + 3];
#pragma unroll
    for (int j = 0; j < 16; ++j) {
        float aj = __shfl(ea, j, 32);
        const float* wr = sWe + j * 128 + c0;
        e0 += aj * wr[0]; e1 += aj * wr[1]; e2 += aj * wr[2]; e3 += aj * wr[3];
    }
    float p = qv.x * (kv.x + e0) + qv.y * (kv.y + e1) +
              qv.z * (kv.z + e2) + qv.w * (kv.w + e3);
    p += __shfl_xor(p, 1, 32);   // reduce HD=16 across lane group of 4
    p += __shfl_xor(p, 2, 32);
    if ((lane & 3) == 0) {
        float sc = p * 0.25f;    // 1/sqrt(HD)
        int h = lane >> 2;
        scores[(size_t)e * 8 + h] = sc;
        atomicMaxF(smax + (size_t)d * 8 + h, sc);
    }
}

__global__ void exp_denom_kernel(float* __restrict__ sc, const float* __restrict__ smax,
                                 const int* __restrict__ ei, float* __restrict__ den, int E) {
    int i = blockIdx.x * blockDim.x + threadIdx.x;
    if (i >= E * 8) return;
    int e = i >> 3, h = i & 7;
    int d = ei[E + e];
    float v = expf(sc[i] - smax[(size_t)d * 8 + h]);
    sc[i] = v;                               // overwrite score with exp (ex)
    atomicAdd(den + (size_t)d * 8 + h, v);
}

__global__ __launch_bounds__(256) void edge_message_kernel(const float* __restrict__ v,
                                                           const float* __restrict__ edge_attr,
                                                           const float* __restrict__ We,
                                                           const float* __restrict__ be,
                                                           const int* __restrict__ ei,
                                                           const float* __restrict__ exs,
                                                           const float* __restrict__ den,
                                                           float* __restrict__ out, int E) {
    __shared__ float sWe[16 * 128];
    __shared__ float sbe[128];
    for (int t = threadIdx.x; t < 2048; t += blockDim.x) sWe[t] = We[t];
    for (int t = threadIdx.x; t < 128; t += blockDim.x) sbe[t] = be[t];
    __syncthreads();
    int lane = threadIdx.x & 31;
    int e = blockIdx.x * 8 + (threadIdx.x >> 5);
    if (e >= E) return;
    int s = ei[e], d = ei[E + e];
    float ea = (lane < 16) ? edge_attr[(size_t)e * 16 + lane] : 0.0f;
    int c0 = lane * 4;
    float4 vv = *(const float4*)(v + (size_t)s * 128 + c0);
    float e0 = sbe[c0], e1 = sbe[c0 + 1], e2 = sbe[c0 + 2], e3 = sbe[c0 + 3];
#pragma unroll
    for (int j = 0; j < 16; ++j) {
        float aj = __shfl(ea, j, 32);
        const float* wr = sWe + j * 128 + c0;
        e0 += aj * wr[0]; e1 += aj * wr[1]; e2 += aj * wr[2]; e3 += aj * wr[3];
    }
    int h = lane >> 2;
    float al = exs[(size_t)e * 8 + h] / fmaxf(den[(size_t)d * 8 + h], 1e-16f);
    float* op = out + (size_t)d * 128 + c0;
    atomicAdd(op + 0, (vv.x + e0) * al);
    atomicAdd(op + 1, (vv.y + e1) * al);
    atomicAdd(op + 2, (vv.z + e2) * al);
    atomicAdd(op + 3, (vv.w + e3) * al);
}

// ---------------- launcher ----------------

extern "C" void kernel_launch(void* const* d_in, const int* in_sizes, int n_in,
                              void* d_out, int out_size, void* d_ws, size_t ws_size,
                              hipStream_t stream) {
    const float* x         = (const float*)d_in[0];
    const float* edge_attr = (const float*)d_in[1];
    const int*   ei        = (const int*)d_in[2];
    const float* Wq = (const float*)d_in[3];  const float* bq = (const float*)d_in[4];
    const float* Wk = (const float*)d_in[5];  const float* bk = (const float*)d_in[6];
    const float* Wv = (const float*)d_in[7];  const float* bv = (const float*)d_in[8];
    const float* Ws = (const float*)d_in[9];  const float* bs = (const float*)d_in[10];
    const float* We = (const float*)d_in[11]; const float* be = (const float*)d_in[12];
    const float* Wp = (const float*)d_in[13]; const float* bp = (const float*)d_in[14];
    const float* ln_a_w = (const float*)d_in[15]; const float* ln_a_b = (const float*)d_in[16];
    const float* ln_m_w = (const float*)d_in[17]; const float* ln_m_b = (const float*)d_in[18];
    const float* W1 = (const float*)d_in[19]; const float* b1 = (const float*)d_in[20];
    const float* W2 = (const float*)d_in[21]; const float* b2 = (const float*)d_in[22];

    const int N = in_sizes[0] / 128;     // 50000
    const int E = in_sizes[1] / 16;      // 800000

    // workspace layout (floats); buffers reused as the pipeline advances
    float* ws   = (float*)d_ws;
    float* xn   = ws;                          // N*128 (LN out; later reused as hln)
    float* qb   = xn   + (size_t)N * 128;      // N*128 (q; later reused as pin)
    float* kb   = qb   + (size_t)N * 128;      // N*128 (k; later reused as out2)
    float* vb   = kb   + (size_t)N * 128;      // N*128
    float* xr   = vb   + (size_t)N * 128;      // N*128
    float* attn = xr   + (size_t)N * 128;      // N*128
    float* sc   = attn + (size_t)N * 128;      // E*8 (scores, then ex)
    float* smax = sc   + (size_t)E * 8;        // N*8
    float* den  = smax + (size_t)N * 8;        // N*8
    float* hid  = den  + (size_t)N * 8;        // N*512

    float* nodes_out = (float*)d_out;
    float* edge_out  = nodes_out + (size_t)N * 128;

    const int lnBlocks = (N + 7) / 8;
    const int rowTiles = N / 16;               // 50000 = 3125*16 exact
    const size_t sh128 = (size_t)16 * 129 * sizeof(float);
    const size_t sh512 = (size_t)16 * 513 * sizeof(float);

    // 1) pre-attention LN
    ln128_kernel<<<lnBlocks, 256, 0, stream>>>(x, ln_a_w, ln_a_b, xn, N);

    // 2) q/k/v/self projections (fp32 WMMA)
    gemm_wmma<0><<<rowTiles, 256, sh128, stream>>>(xn, Wq, bq, nullptr, qb, 128, 128);
    gemm_wmma<0><<<rowTiles, 256, sh128, stream>>>(xn, Wk, bk, nullptr, kb, 128, 128);
    gemm_wmma<0><<<rowTiles, 256, sh128, stream>>>(xn, Wv, bv, nullptr, vb, 128, 128);
    gemm_wmma<0><<<rowTiles, 256, sh128, stream>>>(xn, Ws, bs, nullptr, xr, 128, 128);

    // 3) init softmax state + output accumulator
    fill_kernel<<<(N * 8 + 255) / 256, 256, 0, stream>>>(smax, N * 8, -INFINITY);
    fill_kernel<<<(N * 8 + 255) / 256, 256, 0, stream>>>(den, N * 8, 0.0f);
    fill_kernel<<<(N * 128 + 255) / 256, 256, 0, stream>>>(attn, N * 128, 0.0f);

    // 4) edge attention: scores + segment-max, exp + segment-sum, weighted scatter
    const int eBlocks = (E + 7) / 8;
    edge_score_kernel<<<eBlocks, 256, 0, stream>>>(qb, kb, edge_attr, We, be, ei, sc, smax, E);
    fix_smax_kernel<<<(N * 8 + 255) / 256, 256, 0, stream>>>(smax, N * 8);
    exp_denom_kernel<<<(E * 8 + 255) / 256, 256, 0, stream>>>(sc, smax, ei, den, E);
    edge_message_kernel<<<eBlocks, 256, 0, stream>>>(vb, edge_attr, We, be, ei, sc, den, attn, E);

    // 5) projection + residual:  out2 = (attn + x_r) @ Wp + bp + x
    add2_kernel<<<(N * 128 + 255) / 256, 256, 0, stream>>>(attn, xr, qb, N * 128); // pin -> qb
    gemm_wmma<1><<<rowTiles, 256, sh128, stream>>>(qb, Wp, bp, x, kb, 128, 128);   // out2 -> kb

    // 6) MLP with pre-LN + residual, final GEMM writes straight to d_out
    ln128_kernel<<<lnBlocks, 256, 0, stream>>>(kb, ln_m_w, ln_m_b, xn, N);         // hln -> xn
    gemm_wmma<2><<<rowTiles, 256, sh128, stream>>>(xn, W1, b1, nullptr, hid, 128, 512); // gelu
    gemm_wmma<1><<<rowTiles, 256, sh512, stream>>>(hid, W2, b2, kb, nodes_out, 512, 128);

    // 7) second output of the tuple: pass-through edge_attr
    copy_kernel<<<(E * 16 + 255) / 256, 256, 0, stream>>>(edge_attr, edge_out, E * 16);
}